// DSTAGNNBlock_52999896433220
// MI455X (gfx1250) — compile-verified
//
#include <hip/hip_runtime.h>
#include <hip/hip_bf16.h>
#include <cstdint>

// ---------------- CDNA5 WMMA helpers (fp32, 16x16x4) ----------------
typedef __attribute__((ext_vector_type(2))) float v2f;
typedef __attribute__((ext_vector_type(8))) float v8f;

__device__ __forceinline__ v8f wmma4(v2f a, v2f b, v8f c) {
    // V_WMMA_F32_16X16X4_F32 : D = A(16x4) x B(4x16) + C(16x16), fp32
    return __builtin_amdgcn_wmma_f32_16x16x4_f32(
        /*neg_a=*/false, a, /*neg_b=*/false, b,
        /*c_mod=*/(short)0, c, /*reuse_a=*/false, /*reuse_b=*/false);
}

__device__ __forceinline__ v2f ldv2(const float* p) { return *(const v2f*)p; }

#define LANE_DECOMP int lane = threadIdx.x & 31; int m = lane & 15; int half = lane >> 4;

static constexpr int B_ = 32, N_ = 512, T_ = 12, O_ = 64, H_ = 3, KH_ = 3;
static constexpr int DM_ = 512, DK_ = 32, DV_ = 32;
__device__ __constant__ float SCALE_ = 0.17677669529663687f; // 1/sqrt(32)

// deterministic pseudo-uniform -> logistic noise (gumbel-diff reparam)
__device__ __forceinline__ float logistic_noise(uint32_t idx) {
    uint32_t h = idx * 2654435761u;
    h ^= h >> 16; h *= 2246822519u;
    h ^= h >> 13; h *= 3266489917u;
    h ^= h >> 16;
    float u = (float)(h >> 8) * (1.0f / 16777216.0f);
    u = fminf(fmaxf(u, 1e-7f), 1.0f - 1e-7f);
    return __logf(u) - __logf(1.0f - u);
}

// ============ K0: stage padded copies of x ============
// xT  [B][16][512] : xT[b][t][n]  = x[b][n][0][t], rows t=12..15 zero  (TEmx, padded)
// xpad[B][512][16] : xpad[b][n][t]= x[b][n][0][t], t=12..15 zero
__global__ __launch_bounds__(256) void k_xpad(const float* __restrict__ x,
                                              float* __restrict__ xT,
                                              float* __restrict__ xpad) {
    int idx = blockIdx.x * 256 + threadIdx.x;      // 0 .. 262143
    int t = idx & 15, n = (idx >> 4) & 511, b = idx >> 13;
    float v = (t < 12) ? x[(size_t)(b * 512 + n) * 12 + t] : 0.0f;
    xpad[idx] = v;
    xT[((size_t)b * 16 + t) * 512 + n] = v;
}

// ============ K1: Q/K/V = TEmx @ W  (per b: [16(t,pad)x512] x [512x96]) ============
__global__ __launch_bounds__(256) void k_qkv(const float* __restrict__ xT,
                                             const float* __restrict__ Wq,
                                             const float* __restrict__ Wk,
                                             const float* __restrict__ Wv,
                                             float* __restrict__ Qb, float* __restrict__ Kb,
                                             float* __restrict__ Vb) {
    int b = blockIdx.x;
    int wave = threadIdx.x >> 5;
    int tile = blockIdx.y * 8 + wave;
    if (tile >= 18) return;                 // wave-uniform
    int which = tile / 6, ct = tile % 6, col0 = ct * 16;
    const float* W = which == 0 ? Wq : (which == 1 ? Wk : Wv);
    float* Out = which == 0 ? Qb : (which == 1 ? Kb : Vb);
    LANE_DECOMP
    const float* A = xT + (size_t)b * 16 * 512;     // [16][512], pad rows zero
    v8f acc = {};
#pragma unroll 2
    for (int k0 = 0; k0 < 512; k0 += 4) {
        int ka = k0 + 2 * half;
        v2f a = ldv2(&A[m * 512 + ka]);                       // contiguous pair
        v2f bb = { W[ka * 96 + col0 + m], W[(ka + 1) * 96 + col0 + m] };
        acc = wmma4(a, bb, acc);
    }
    float* o = Out + b * 16 * 96;
#pragma unroll
    for (int r = 0; r < 8; ++r) o[(r + 8 * half) * 96 + col0 + m] = acc[r]; // rows>=12 exact 0
}

// ============ K2: scores (+res_att) -> out, softmax, ctx = attn@V ============
__global__ __launch_bounds__(256) void k_attn(const float* __restrict__ Qb,
                                              const float* __restrict__ Kb,
                                              const float* __restrict__ Vb,
                                              const float* __restrict__ res_att,
                                              float* __restrict__ scoresOut,
                                              float* __restrict__ ctx) {
    int bh = blockIdx.x; int b = bh / H_, h = bh % H_;
    int tid = threadIdx.x;
    __shared__ float sc[12][12];
    __shared__ float at[12][12];
    const float* Q = Qb + b * 16 * 96 + h * 32;
    const float* K = Kb + b * 16 * 96 + h * 32;
    const float* V = Vb + b * 16 * 96 + h * 32;
    for (int idx = tid; idx < 144; idx += 256) {
        int q = idx / 12, p = idx % 12;
        float s = 0.0f;
        for (int d = 0; d < 32; ++d) s += Q[q * 96 + d] * K[p * 96 + d];
        s = s * SCALE_ + res_att[(size_t)bh * 144 + q * 12 + p];
        sc[q][p] = s;
        scoresOut[(size_t)bh * 144 + q * 12 + p] = s;   // re_At output
    }
    __syncthreads();
    if (tid < 12) {
        int q = tid;
        float mx = sc[q][0];
        for (int p = 1; p < 12; ++p) mx = fmaxf(mx, sc[q][p]);
        float e[12], sum = 0.0f;
        for (int p = 0; p < 12; ++p) { e[p] = __expf(sc[q][p] - mx); sum += e[p]; }
        float inv = 1.0f / sum;
        for (int p = 0; p < 12; ++p) at[q][p] = e[p] * inv;
    }
    __syncthreads();
    float* C = ctx + b * 16 * 96 + h * 32;
    for (int idx = tid; idx < 12 * 32; idx += 256) {
        int q = idx >> 5, v = idx & 31;
        float s = 0.0f;
        for (int p = 0; p < 12; ++p) s += at[q][p] * V[p * 96 + v];
        C[q * 96 + v] = s;
    }
    for (int idx = tid; idx < 4 * 32; idx += 256) {   // zero pad rows 12..15
        int rr = 12 + (idx >> 5), v = idx & 31;
        C[rr * 96 + v] = 0.0f;
    }
}

// ============ K3: TAT = tanh(ctx @ Wfc + bfc) + TEmx   ([16x96]x[96x512] per b) ============
__global__ __launch_bounds__(256) void k_tat(const float* __restrict__ ctx,
                                             const float* __restrict__ Wfc,
                                             const float* __restrict__ bfc,
                                             const float* __restrict__ x,
                                             float* __restrict__ TAT) {
    int b = blockIdx.x;
    int wave = threadIdx.x >> 5;
    int col0 = (blockIdx.y * 8 + wave) * 16;  // 32 col tiles
    LANE_DECOMP
    const float* A = ctx + b * 16 * 96;
    v8f acc = {};
#pragma unroll 2
    for (int k0 = 0; k0 < 96; k0 += 4) {
        int ka = k0 + 2 * half;
        v2f a = ldv2(&A[m * 96 + ka]);                        // contiguous pair
        v2f bb = { Wfc[ka * 512 + col0 + m], Wfc[(ka + 1) * 512 + col0 + m] };
        acc = wmma4(a, bb, acc);
    }
    float* o = TAT + b * 16 * 512;
    const float* xb = x + b * N_ * T_;
#pragma unroll
    for (int r = 0; r < 8; ++r) {
        int row = r + 8 * half, col = col0 + m;
        float v = 0.0f;
        if (row < 12) v = tanhf(acc[r] + bfc[col]) + xb[col * T_ + row];
        o[row * 512 + col] = v;
    }
}

// ============ K4: x_TAt = TAT^T @ Wpre + bpre   (per b: [512x12]x[12x512]) ============
__global__ __launch_bounds__(256) void k_xtat(const float* __restrict__ TAT,
                                              const float* __restrict__ Wpre,
                                              const float* __restrict__ bpre,
                                              float* __restrict__ xTAt) {
    int b = blockIdx.x;
    int wave = threadIdx.x >> 5;
    int tile = blockIdx.y * 8 + wave;     // 1024 tiles
    int row0 = (tile >> 5) * 16, col0 = (tile & 31) * 16;
    LANE_DECOMP
    const float* TA = TAT + b * 16 * 512;
    v8f acc = {};
#pragma unroll
    for (int k0 = 0; k0 < 12; k0 += 4) {
        int ka = k0 + 2 * half;
        v2f a = { TA[ka * 512 + row0 + m], TA[(ka + 1) * 512 + row0 + m] }; // A[n][t]=TAT[t][n]
        v2f bb = { Wpre[ka * 512 + col0 + m], Wpre[(ka + 1) * 512 + col0 + m] };
        acc = wmma4(a, bb, acc);
    }
    float* o = xTAt + (size_t)b * 512 * 512;
#pragma unroll
    for (int r = 0; r < 8; ++r) {
        int row = row0 + r + 8 * half, col = col0 + m;
        o[row * 512 + col] = acc[r] + bpre[col];
    }
}

// ============ K5: Qs / Ks fused  ([16384x512]x[512x96], Ks stored transposed) ============
__global__ __launch_bounds__(256) void k_qsks(const float* __restrict__ xTAt,
                                              const float* __restrict__ Wqs,
                                              const float* __restrict__ Wks,
                                              float* __restrict__ Qs,
                                              float* __restrict__ KsT) {
    int wave = threadIdx.x >> 5;
    int tile = blockIdx.x * 8 + wave;     // 6144 tiles
    int mt = tile / 6, ct = tile % 6;
    int r0 = mt * 16, b = r0 >> 9, n0 = r0 & 511, col0 = ct * 16;
    LANE_DECOMP
    const float* A = xTAt + (size_t)r0 * 512;
    v8f cq = {}, ck = {};
#pragma unroll 2
    for (int k0 = 0; k0 < 512; k0 += 4) {
        int ka = k0 + 2 * half;
        v2f a = ldv2(&A[m * 512 + ka]);                       // contiguous pair
        v2f bq = { Wqs[ka * 96 + col0 + m], Wqs[(ka + 1) * 96 + col0 + m] };
        v2f bk = { Wks[ka * 96 + col0 + m], Wks[(ka + 1) * 96 + col0 + m] };
        cq = wmma4(a, bq, cq);
        ck = wmma4(a, bk, ck);
    }
    int col = col0 + m, kh = col >> 5, d = col & 31;
#pragma unroll
    for (int r = 0; r < 8; ++r) {
        int row = n0 + r + 8 * half;
        Qs[((size_t)(b * 3 + kh) * 512 + row) * 32 + d] = cq[r];
        KsT[((size_t)(b * 3 + kh) * 32 + d) * 512 + row] = ck[r];
    }
}

// ============ K6: fused  a = softmax(Qs@Ks^T*scale + noise)  + KL partials ============
// One block owns a 16-row x 512-col strip of one (b,k) head: 8 waves x 4 col-tiles
// of WMMA, stage into LDS, row-softmax via 16-lane shuffles, single write of `a`.
__global__ __launch_bounds__(256) void k_logits_softmax(const float* __restrict__ Qs,
                                                        const float* __restrict__ KsT,
                                                        float* __restrict__ abuf,
                                                        float* __restrict__ klpart) {
    constexpr int LS = 540;                       // row stride pad: kills LDS bank conflicts
    __shared__ float lds[16 * LS];
    __shared__ float red[256];
    int blk = blockIdx.x;                         // bk*32 + rowblock
    int bk = blk >> 5, row0 = (blk & 31) * 16;
    int tid = threadIdx.x;
    int wave = tid >> 5;
    LANE_DECOMP
    const float* Aq = Qs + (size_t)bk * 512 * 32 + (size_t)(row0 + m) * 32;
    const float* Bk = KsT + (size_t)bk * 32 * 512;
    int c0 = (wave * 4 + 0) * 16 + m;
    int c1 = (wave * 4 + 1) * 16 + m;
    int c2 = (wave * 4 + 2) * 16 + m;
    int c3 = (wave * 4 + 3) * 16 + m;
    v8f a0 = {}, a1 = {}, a2 = {}, a3 = {};
#pragma unroll
    for (int k0 = 0; k0 < 32; k0 += 4) {
        int ka = k0 + 2 * half;
        v2f a = ldv2(&Aq[ka]);                    // contiguous pair, shared by 4 tiles
        v2f b0 = { Bk[ka * 512 + c0], Bk[(ka + 1) * 512 + c0] };
        v2f b1 = { Bk[ka * 512 + c1], Bk[(ka + 1) * 512 + c1] };
        v2f b2 = { Bk[ka * 512 + c2], Bk[(ka + 1) * 512 + c2] };
        v2f b3 = { Bk[ka * 512 + c3], Bk[(ka + 1) * 512 + c3] };
        a0 = wmma4(a, b0, a0);
        a1 = wmma4(a, b1, a1);
        a2 = wmma4(a, b2, a2);
        a3 = wmma4(a, b3, a3);
    }
    uint32_t base = (uint32_t)(bk * 262144 + row0 * 512);
#pragma unroll
    for (int r = 0; r < 8; ++r) {
        int row = r + 8 * half;
        uint32_t rb = base + (uint32_t)row * 512;
        lds[row * LS + c0] = a0[r] * SCALE_ + logistic_noise(rb + c0);
        lds[row * LS + c1] = a1[r] * SCALE_ + logistic_noise(rb + c1);
        lds[row * LS + c2] = a2[r] * SCALE_ + logistic_noise(rb + c2);
        lds[row * LS + c3] = a3[r] * SCALE_ + logistic_noise(rb + c3);
    }
    __syncthreads();
    // ---- row softmax: row = tid>>4, 16 lanes per row, 32 cols per lane ----
    int srow = tid >> 4, sl = tid & 15;
    float vbuf[32];
    float mx = -3.0e38f;
#pragma unroll
    for (int j = 0; j < 32; ++j) {
        float v = lds[srow * LS + sl + j * 16];
        vbuf[j] = v;
        mx = fmaxf(mx, v);
    }
#pragma unroll
    for (int d = 1; d < 16; d <<= 1) mx = fmaxf(mx, __shfl_xor(mx, d, 16));
    float sum = 0.0f;
#pragma unroll
    for (int j = 0; j < 32; ++j) { float e = __expf(vbuf[j] - mx); vbuf[j] = e; sum += e; }
#pragma unroll
    for (int d = 1; d < 16; d <<= 1) sum += __shfl_xor(sum, d, 16);
    float inv = 1.0f / sum;
    float* arow = abuf + (size_t)bk * 262144 + (size_t)(row0 + srow) * 512;
    const float rinv = 1.0f / 0.9f, qf = 1.0f / 0.100001f;   // 1/r , 1/(1-r+1e-6)
    float kl = 0.0f;
#pragma unroll
    for (int j = 0; j < 32; ++j) {
        float av = vbuf[j] * inv;
        arow[sl + j * 16] = av;
        kl += av * __logf(av * rinv + 1e-6f) + (1.0f - av) * __logf((1.0f - av) * qf + 1e-6f);
    }
    red[tid] = kl;
    __syncthreads();
    for (int st = 128; st > 0; st >>= 1) { if (tid < st) red[tid] += red[tid + st]; __syncthreads(); }
    if (tid == 0) klpart[blk] = red[0];
}

__global__ __launch_bounds__(256) void k_kl_reduce(const float* __restrict__ klpart,
                                                   float* __restrict__ outKl) {
    __shared__ float red[256];
    int tid = threadIdx.x;
    float s = 0.0f;
    for (int i = tid; i < 3072; i += 256) s += klpart[i];
    red[tid] = s;
    __syncthreads();
    for (int st = 128; st > 0; st >>= 1) { if (tid < st) red[tid] += red[tid + st]; __syncthreads(); }
    if (tid == 0) *outKl = red[0] / 25165824.0f;   // B*K*N*N
}

// ============ K9: S = a^T @ x   (per b,k: [512x512]x[512x12->16]) ============
__global__ __launch_bounds__(256) void k_sgemm(const float* __restrict__ a_mat,
                                               const float* __restrict__ xpad,
                                               float* __restrict__ S) {
    int wave = threadIdx.x >> 5;
    int tile = blockIdx.x * 8 + wave;     // 3072 tiles
    int bk = tile >> 5, row0 = (tile & 31) * 16;
    int b = bk / 3;
    LANE_DECOMP
    const float* A = a_mat + (size_t)bk * 262144;
    const float* xb = xpad + (size_t)b * 512 * 16;   // [512][16], t pad zero
    v8f acc = {};
#pragma unroll 2
    for (int k0 = 0; k0 < 512; k0 += 4) {
        int ka = k0 + 2 * half;
        v2f a = { A[ka * 512 + row0 + m], A[(ka + 1) * 512 + row0 + m] };  // a[m_node][n']
        v2f bb = { xb[ka * 16 + m], xb[(ka + 1) * 16 + m] };               // unconditional
        acc = wmma4(a, bb, acc);
    }
#pragma unroll
    for (int r = 0; r < 8; ++r) {
        int row = row0 + r + 8 * half;
        if (m < 12) S[((size_t)bk * 512 + row) * 12 + m] = acc[r];
    }
}

// ============ K10: C[k][o] = sum_o' Theta[k,o'] * Wlin[o'*K+k, o]  (collapsed Wlin) ============
__global__ __launch_bounds__(256) void k_cbuf(const float* __restrict__ Theta,
                                              const float* __restrict__ Wlin,
                                              float* __restrict__ Cb) {
    int idx = threadIdx.x;
    if (idx >= 192) return;
    int k = idx / 64, o = idx % 64;
    float s = 0.0f;
    for (int op = 0; op < 64; ++op) s += Theta[k * 64 + op] * Wlin[(op * 3 + k) * 64 + o];
    Cb[k * 64 + o] = s;
}

// ============ K11: fused gcn / time-conv / residual epilogue ============
__global__ __launch_bounds__(64) void k_final(const float* __restrict__ S,
                                              const float* __restrict__ Cb,
                                              const float* __restrict__ blin,
                                              const float* __restrict__ Wfcmy,
                                              const float* __restrict__ bfcmy,
                                              const float* __restrict__ Wres,
                                              const float* __restrict__ bres,
                                              const float* __restrict__ x,
                                              float* __restrict__ out) {
    int bn = blockIdx.x; int b = bn >> 9, n = bn & 511;
    int o = threadIdx.x;
    __shared__ float g[12][64];
    float c0 = Cb[o], c1 = Cb[64 + o], c2 = Cb[128 + o], bl = blin[o];
    const float* S0 = S + ((size_t)(b * 3 + 0) * 512 + n) * 12;
    const float* S1 = S + ((size_t)(b * 3 + 1) * 512 + n) * 12;
    const float* S2 = S + ((size_t)(b * 3 + 2) * 512 + n) * 12;
#pragma unroll
    for (int t = 0; t < 12; ++t)
        g[t][o] = tanhf(S0[t] * c0 + S1[t] * c1 + S2[t] * c2 + bl);
    __syncthreads();
    const float* xv = x + (size_t)bn * 12;
    float wr = Wres[o], br = bres[o];
    float* op = out + ((size_t)bn * 64 + o) * 12;
#pragma unroll
    for (int t = 0; t < 12; ++t) {
        float tc = bfcmy[t];
#pragma unroll
        for (int tp = 0; tp < 12; ++tp) tc += g[tp][o] * Wfcmy[tp * 12 + t];
        op[t] = tanhf(xv[t] * wr + br + tanhf(tc));
    }
}

// ============================== launch ==============================
extern "C" void kernel_launch(void* const* d_in, const int* in_sizes, int n_in,
                              void* d_out, int out_size, void* d_ws, size_t ws_size,
                              hipStream_t stream) {
    const float* x       = (const float*)d_in[0];
    const float* res_att = (const float*)d_in[1];
    const float* Wq      = (const float*)d_in[2];
    const float* Wk      = (const float*)d_in[3];
    const float* Wv      = (const float*)d_in[4];
    const float* Wfc     = (const float*)d_in[5];
    const float* bfc     = (const float*)d_in[6];
    const float* Wpre    = (const float*)d_in[7];
    const float* bpre    = (const float*)d_in[8];
    const float* Wqs     = (const float*)d_in[9];
    const float* Wks     = (const float*)d_in[10];
    const float* Theta   = (const float*)d_in[11];
    const float* Wlin    = (const float*)d_in[12];
    const float* blin    = (const float*)d_in[13];
    const float* Wfcmy   = (const float*)d_in[14];
    const float* bfcmy   = (const float*)d_in[15];
    const float* Wres    = (const float*)d_in[16];
    const float* bres    = (const float*)d_in[17];

    float* ws = (float*)d_ws;
    size_t off = 0;
    float* xT     = ws + off; off += (size_t)32 * 16 * 512;         // 262144
    float* xpad   = ws + off; off += (size_t)32 * 512 * 16;         // 262144
    float* Qb     = ws + off; off += (size_t)32 * 16 * 96;          // 49152
    float* Kb     = ws + off; off += (size_t)32 * 16 * 96;
    float* Vb     = ws + off; off += (size_t)32 * 16 * 96;
    float* ctx    = ws + off; off += (size_t)32 * 16 * 96;
    float* TAT    = ws + off; off += (size_t)32 * 16 * 512;         // 262144
    float* xTAt   = ws + off; off += (size_t)32 * 512 * 512;        // 8388608
    float* Qs     = ws + off; off += (size_t)32 * 3 * 512 * 32;     // 1572864
    float* KsT    = ws + off; off += (size_t)32 * 3 * 32 * 512;     // 1572864
    float* abuf   = ws + off; off += (size_t)32 * 3 * 512 * 512;    // 25165824
    float* Sbuf   = ws + off; off += (size_t)32 * 3 * 512 * 12;     // 589824
    float* klpart = ws + off; off += (size_t)3072;
    float* Cb     = ws + off; off += (size_t)192;

    float* out_xres   = (float*)d_out;                           // [B,N,O,T]
    float* out_scores = out_xres + (size_t)32 * 512 * 64 * 12;   // [B,1,H,T,T]
    float* out_kl     = out_scores + (size_t)32 * 3 * 12 * 12;   // scalar

    k_xpad<<<1024, 256, 0, stream>>>(x, xT, xpad);
    k_qkv<<<dim3(32, 3), 256, 0, stream>>>(xT, Wq, Wk, Wv, Qb, Kb, Vb);
    k_attn<<<96, 256, 0, stream>>>(Qb, Kb, Vb, res_att, out_scores, ctx);
    k_tat<<<dim3(32, 4), 256, 0, stream>>>(ctx, Wfc, bfc, x, TAT);
    k_xtat<<<dim3(32, 128), 256, 0, stream>>>(TAT, Wpre, bpre, xTAt);
    k_qsks<<<768, 256, 0, stream>>>(xTAt, Wqs, Wks, Qs, KsT);
    k_logits_softmax<<<3072, 256, 0, stream>>>(Qs, KsT, abuf, klpart);
    k_kl_reduce<<<1, 256, 0, stream>>>(klpart, out_kl);
    k_cbuf<<<1, 256, 0, stream>>>(Theta, Wlin, Cb);
    k_sgemm<<<384, 256, 0, stream>>>(abuf, xpad, Sbuf);
    k_final<<<16384, 64, 0, stream>>>(Sbuf, Cb, blin, Wfcmy, bfcmy, Wres, bres, x, out_xres);
}